// Encoder_66082366816650
// MI455X (gfx1250) — compile-verified
//
#include <hip/hip_runtime.h>
#include <hip/hip_bf16.h>

// Fused 3-stage strided conv1d encoder for MI455X (gfx1250).
//
// Roofline: ~290 MB mandatory HBM traffic @ 23.3 TB/s => ~12.5 us floor;
// ~0.7 GFLOP of math => bandwidth-bound. x streams through the CDNA5
// async global->LDS path (ASYNCcnt, th:TH_LOAD_NT since the 256 MB stream
// exceeds the 192 MB L2); y1/y2 live in LDS; output leaves via NT stores.
// LDS phases are pair-vectorized (2 outputs / thread iteration) so each
// pair costs 2x ds_load_b128 + 1x ds_store_b64 instead of 12x ds_load_b32
// + 2x ds_store_b32 — keeps LDS issue rate below the HBM stream time.

#define TPB     256
#define KW      6
#define L_IN    16384
#define L3      2044
#define TILE_N  1024   // final outputs per tile (2 tiles per row)

typedef float v2f __attribute__((ext_vector_type(2)));  // clang vector: OK for NT builtin

// Two outputs from two aligned float4 windows: out0 uses a.x..b.y, out1 uses a.z..b.w
__device__ __forceinline__ void conv_pair(const float* __restrict__ W, float B,
                                          float4 a, float4 b,
                                          float& r0, float& r1) {
    float t0 = B, t1 = B;
    t0 = fmaf(W[0], a.x, t0); t1 = fmaf(W[0], a.z, t1);
    t0 = fmaf(W[1], a.y, t0); t1 = fmaf(W[1], a.w, t1);
    t0 = fmaf(W[2], a.z, t0); t1 = fmaf(W[2], b.x, t1);
    t0 = fmaf(W[3], a.w, t0); t1 = fmaf(W[3], b.y, t1);
    t0 = fmaf(W[4], b.x, t0); t1 = fmaf(W[4], b.z, t1);
    t0 = fmaf(W[5], b.y, t0); t1 = fmaf(W[5], b.w, t1);
    r0 = t0; r1 = t1;
}

__global__ __launch_bounds__(TPB) void encoder_fused_kernel(
    const float* __restrict__ x,
    const float* __restrict__ w1, const float* __restrict__ b1,
    const float* __restrict__ w2, const float* __restrict__ b2,
    const float* __restrict__ w3, const float* __restrict__ b3,
    float* __restrict__ out)
{
    // Max sizes for n = TILE_N: x: 8n+28 = 8220, y1: 4n+12 = 4108, y2: 2n+4 = 2052
    __shared__ alignas(16) float sX [8224];   // padded: covers float4 read at x[8n+24..8n+27]
    __shared__ alignas(16) float sY1[4112];
    __shared__ alignas(16) float sY2[2056];

    const int bid  = blockIdx.x;
    const int row  = bid >> 1;
    const int tile = bid & 1;
    const int tid  = threadIdx.x;

    const int j0 = tile * TILE_N;
    const int n  = min(TILE_N, L3 - j0);   // 1024 or 1020 (always even)
    const int m2 = 2 * n + 4;              // y2 elements needed (even)
    const int m1 = 4 * n + 12;             // y1 elements needed (even)
    const int cx = 8 * n + 28;             // x elements needed (tile start 32B aligned)
    const int c4 = (cx + 3) >> 2;          // float4 chunks (exact fit at row end)

    const float* xrow = x + (size_t)row * L_IN + 8 * j0;

    // ---- Phase 0: async copy x tile into LDS (GLOBAL_LOAD_ASYNC_TO_LDS_B128, NT) ----
    const unsigned lds_base = (unsigned)(size_t)(&sX[0]); // LDS offset = low 32 bits of flat address
    for (int c = tid; c < c4; c += TPB) {
        unsigned           lofs = lds_base + (unsigned)c * 16u;
        unsigned long long gptr = (unsigned long long)(xrow + c * 4);
        asm volatile("global_load_async_to_lds_b128 %0, %1, off th:TH_LOAD_NT"
                     :: "v"(lofs), "v"(gptr) : "memory");
    }

    // Load taps/biases while the async DMA is in flight (uniform -> s_load).
    float W1[KW], W2[KW], W3[KW];
    #pragma unroll
    for (int k = 0; k < KW; ++k) { W1[k] = w1[k]; W2[k] = w2[k]; W3[k] = w3[k]; }
    const float B1 = b1[0], B2 = b2[0], B3 = b3[0];

    asm volatile("s_wait_asynccnt 0" ::: "memory"); // this wave's async copies done
    __syncthreads();                                // all waves' tiles visible

    const float4* x4  = reinterpret_cast<const float4*>(sX);
    const float4* y14 = reinterpret_cast<const float4*>(sY1);
    const float4* y24 = reinterpret_cast<const float4*>(sY2);
    float2*       y1v = reinterpret_cast<float2*>(sY1);
    float2*       y2v = reinterpret_cast<float2*>(sY2);

    // ---- Phase 1: conv1 (x -> y1), 2 outputs per iter: ds_load_b128 x2 + ds_store_b64 ----
    for (int p = tid; p < (m1 >> 1); p += TPB) {
        float4 a = x4[p], b = x4[p + 1];
        float r0, r1;
        conv_pair(W1, B1, a, b, r0, r1);
        y1v[p] = make_float2(r0, r1);
    }
    __syncthreads();

    // ---- Phase 2: conv2 (y1 -> y2) ----
    for (int p = tid; p < (m2 >> 1); p += TPB) {
        float4 a = y14[p], b = y14[p + 1];
        float r0, r1;
        conv_pair(W2, B2, a, b, r0, r1);
        y2v[p] = make_float2(r0, r1);
    }
    __syncthreads();

    // ---- Phase 3: conv3 (y2 -> out), coalesced 8B non-temporal global stores ----
    v2f* orow = reinterpret_cast<v2f*>(out + (size_t)row * L3 + j0); // 8B aligned
    for (int p = tid; p < (n >> 1); p += TPB) {
        float4 a = y24[p], b = y24[p + 1];
        float r0, r1;
        conv_pair(W3, B3, a, b, r0, r1);
        v2f v; v.x = r0; v.y = r1;
        __builtin_nontemporal_store(v, &orow[p]);
    }
}

extern "C" void kernel_launch(void* const* d_in, const int* in_sizes, int n_in,
                              void* d_out, int out_size, void* d_ws, size_t ws_size,
                              hipStream_t stream) {
    const float* x  = (const float*)d_in[0];
    const float* w1 = (const float*)d_in[1];
    const float* b1 = (const float*)d_in[2];
    const float* w2 = (const float*)d_in[3];
    const float* b2 = (const float*)d_in[4];
    const float* w3 = (const float*)d_in[5];
    const float* b3 = (const float*)d_in[6];
    float* out = (float*)d_out;

    const int rows = in_sizes[0] / L_IN;   // 4096
    dim3 grid(rows * 2), block(TPB);
    encoder_fused_kernel<<<grid, block, 0, stream>>>(x, w1, b1, w2, b2, w3, b3, out);
}